// AMTCL_77867757077077
// MI455X (gfx1250) — compile-verified
//
#include <hip/hip_runtime.h>
#include <hip/hip_bf16.h>

typedef __attribute__((ext_vector_type(2))) float v2f;
typedef __attribute__((ext_vector_type(8))) float v8f;

#define NCLS   100
#define CPAD   112        // 7 tiles of 16 classes
#define NDIM   384
#define NBATCH 4096
#define NBLK   (NBATCH / 16)   // 256 row-tiles

// ---------------------------------------------------------------------------
// Kernel 1: per-class matrices  M1[k,d] = 2^w * c,  M2[k,d] = 2^w,
//           tcc[k] = sum_d 2^w * c^2.  Pad rows (k>=100) zeroed.
// ---------------------------------------------------------------------------
__global__ void AMTCL_prep_classes(const float* __restrict__ centers,
                                   const float* __restrict__ cw,
                                   float* __restrict__ M1, float* __restrict__ M2,
                                   float* __restrict__ tcc) {
    const int k = blockIdx.x;      // 0..111
    const int d = threadIdx.x;     // 0..383
    __shared__ float red[NDIM];
    float val = 0.0f;
    if (k < NCLS) {
        float c = centers[k * NDIM + d];
        float w = exp2f(cw[k * NDIM + d]);
        M1[k * NDIM + d] = w * c;
        M2[k * NDIM + d] = w;
        val = w * c * c;
    } else {
        M1[k * NDIM + d] = 0.0f;
        M2[k * NDIM + d] = 0.0f;
    }
    red[d] = val;
    __syncthreads();
    for (int s = 192; s >= 6; s >>= 1) {          // 384->192->...->6
        if (d < s) red[d] += red[d + s];
        __syncthreads();
    }
    if (d == 0) {
        float t = 0.0f;
        for (int i = 0; i < 6; ++i) t += red[i];
        tcc[k] = t;
    }
}

// ---------------------------------------------------------------------------
// Kernel 2: centers_dist[i] = sqrt(min_{j!=i} sum_d 2^w[i,d]*(c[i,d]-c[j,d])^2)
// (second-smallest of the sorted row; diagonal is exactly 0)
// ---------------------------------------------------------------------------
__global__ void AMTCL_centers_dist(const float* __restrict__ centers,
                                   const float* __restrict__ cw,
                                   float* __restrict__ cdist) {
    const int i   = blockIdx.x;    // 0..99
    const int tid = threadIdx.x;   // 0..127
    __shared__ float ci[NDIM];
    __shared__ float wi[NDIM];
    for (int d = tid; d < NDIM; d += 128) {
        ci[d] = centers[i * NDIM + d];
        wi[d] = exp2f(cw[i * NDIM + d]);
    }
    __syncthreads();
    float best = 3.0e38f;
    for (int j = tid; j < NCLS; j += 128) {
        if (j == i) continue;
        const float* cj = centers + (size_t)j * NDIM;
        float s = 0.0f;
        for (int d = 0; d < NDIM; ++d) {
            float diff = ci[d] - cj[d];
            s = fmaf(wi[d] * diff, diff, s);
        }
        best = fminf(best, s);
    }
    __shared__ float red[128];
    red[tid] = best;
    __syncthreads();
    for (int st = 64; st > 0; st >>= 1) {
        if (tid < st) red[tid] = fminf(red[tid], red[tid + st]);
        __syncthreads();
    }
    if (tid == 0) cdist[i] = sqrtf(red[0]);
}

// ---------------------------------------------------------------------------
// Kernel 3: class presence flags (deterministic, no atomics, no pre-zeroing)
// ---------------------------------------------------------------------------
__global__ void AMTCL_presence(const int* __restrict__ targets,
                               int* __restrict__ present) {
    const int k = threadIdx.x;
    if (k >= CPAD) return;
    int flag = 0;
    for (int j = 0; j < NBATCH; ++j) flag |= (targets[j] == k);
    present[k] = flag;
}

// ---------------------------------------------------------------------------
// Kernel 4: fused [B,C] WMMA GEMMs + distance + hardest-pos/neg reductions.
// One wave32 per 16-row tile. Two v_wmma_f32_16x16x4_f32 per K-step.
// ---------------------------------------------------------------------------
__global__ __launch_bounds__(32)
void AMTCL_main_wmma(const float* __restrict__ inputs,
                     const int* __restrict__ targets,
                     const float* __restrict__ M1, const float* __restrict__ M2,
                     const float* __restrict__ tcc, const float* __restrict__ cdist,
                     const int* __restrict__ present,
                     float* __restrict__ partials) {
    const int lane = threadIdx.x;        // 0..31
    const int half = lane >> 4;          // 0 or 1
    const int lp   = lane & 15;
    const int m0   = blockIdx.x * 16;

    // targets for the 8 rows this lane's C/D registers cover: M = r + 8*half
    int trow[8];
#pragma unroll
    for (int r = 0; r < 8; ++r) trow[r] = targets[m0 + half * 8 + r];

    float min_an[8], dself[8];
#pragma unroll
    for (int r = 0; r < 8; ++r) { min_an[r] = 3.0e38f; dself[r] = 0.0f; }

    // A-matrix per-lane base: row = m0+lp, K-pair = 2*half  (ISA 16x4 f32 layout)
    const float* arow = inputs + (size_t)(m0 + lp) * NDIM + half * 2;

    for (int nt = 0; nt < CPAD / 16; ++nt) {
        const int n0 = nt * 16;
        // B-matrix per-lane base: col (class) = n0+lp, K-pair = 2*half
        const float* b1 = M1 + (size_t)(n0 + lp) * NDIM + half * 2;
        const float* b2 = M2 + (size_t)(n0 + lp) * NDIM + half * 2;

        v8f xc = {};   // inputs   @ (w*c)^T  tile
        v8f xx = {};   // inputs^2 @  w^T     tile
        for (int k0 = 0; k0 < NDIM; k0 += 4) {
            v2f a  = *(const v2f*)(arow + k0);
            v2f a2 = a * a;
            v2f w1 = *(const v2f*)(b1 + k0);
            v2f w2 = *(const v2f*)(b2 + k0);
            xc = __builtin_amdgcn_wmma_f32_16x16x4_f32(false, a,  false, w1,
                                                       (short)0, xc, false, false);
            xx = __builtin_amdgcn_wmma_f32_16x16x4_f32(false, a2, false, w2,
                                                       (short)0, xx, false, false);
        }

        const int  cls   = n0 + lp;
        const bool valid = (cls < NCLS);
        const float tc   = valid ? tcc[cls] : 0.0f;
        const bool pres  = valid && (present[cls] != 0);
#pragma unroll
        for (int r = 0; r < 8; ++r) {
            float d2   = tc - 2.0f * xc[r] + xx[r];
            float dist = sqrtf(fmaxf(d2, 1e-12f));
            if (valid) {
                if (cls == trow[r])      dself[r]  = dist;           // hardest positive
                else if (pres)           min_an[r] = fminf(min_an[r], dist); // hardest negative
            }
        }
    }

    // butterfly reductions across the 16 lanes of each half (wave32)
#pragma unroll
    for (int r = 0; r < 8; ++r) {
        for (int off = 8; off >= 1; off >>= 1) {
            min_an[r] = fminf(min_an[r], __shfl_xor(min_an[r], off, 16));
            dself[r]  = fmaxf(dself[r],  __shfl_xor(dself[r],  off, 16));
        }
    }

    float acc = 0.0f;
#pragma unroll
    for (int r = 0; r < 8; ++r) {
        float cc = cdist[trow[r]];
        float an = min_an[r];
        acc += dself[r] + ((an >= cc) ? 0.0f : cc - an);
    }
    acc += __shfl_xor(acc, 16, 32);      // combine the two 8-row halves
    if (lane == 0) partials[blockIdx.x] = acc;
}

// ---------------------------------------------------------------------------
// Kernel 5: deterministic fixed-order final reduction
// ---------------------------------------------------------------------------
__global__ void AMTCL_final_reduce(const float* __restrict__ partials,
                                   float* __restrict__ out) {
    __shared__ float red[NBLK];
    const int t = threadIdx.x;
    red[t] = partials[t];
    __syncthreads();
    for (int s = NBLK / 2; s > 0; s >>= 1) {
        if (t < s) red[t] += red[t + s];
        __syncthreads();
    }
    if (t == 0) out[0] = red[0] / (float)NBATCH;
}

// ---------------------------------------------------------------------------
extern "C" void kernel_launch(void* const* d_in, const int* in_sizes, int n_in,
                              void* d_out, int out_size, void* d_ws, size_t ws_size,
                              hipStream_t stream) {
    const float* inputs  = (const float*)d_in[0];   // [4096,384] f32
    const int*   targets = (const int*)  d_in[1];   // [4096] i32
    // d_in[2] = epoch_number (unused)
    const float* centers = (const float*)d_in[3];   // [100,384] f32
    const float* cw      = (const float*)d_in[4];   // [100,384] f32
    float* out = (float*)d_out;

    // workspace layout (bytes)
    char* ws = (char*)d_ws;
    float* M1       = (float*)(ws);                                 // 112*384 f32
    float* M2       = (float*)(ws + (size_t)CPAD * NDIM * 4);       // 112*384 f32
    float* tcc      = (float*)(ws + (size_t)2 * CPAD * NDIM * 4);   // 112 f32
    float* cdist    = tcc + CPAD;                                   // 100 f32
    int*   present  = (int*)(cdist + NCLS + 4);                     // 112 i32
    float* partials = (float*)(present + CPAD + 4);                 // 256 f32

    AMTCL_prep_classes<<<CPAD, NDIM, 0, stream>>>(centers, cw, M1, M2, tcc);
    AMTCL_centers_dist<<<NCLS, 128, 0, stream>>>(centers, cw, cdist);
    AMTCL_presence<<<1, 128, 0, stream>>>(targets, present);
    AMTCL_main_wmma<<<NBLK, 32, 0, stream>>>(inputs, targets, M1, M2, tcc,
                                             cdist, present, partials);
    AMTCL_final_reduce<<<1, NBLK, 0, stream>>>(partials, out);
}